// GNN_37340445671691
// MI455X (gfx1250) — compile-verified
//
#include <hip/hip_runtime.h>
#include <math.h>

// Problem constants
#define NB   512
#define NN   70
#define FF   42
#define F1C  32
#define DD   50
#define EE   (NB*NN*NN)      // 2,508,800 edge rows
#define BNR  (NB*NN)         // 35,840 node rows
#define KXM  82              // F1 + D
#define EPS  1e-5f

typedef __attribute__((ext_vector_type(2))) float v2f;
typedef __attribute__((ext_vector_type(8))) float v8f;

// ---------------------------------------------------------------------------
// Generic WMMA GEMM: Out[M,C] = act(A[M,K] @ W[K,C]); optional (n,m) row remap
// on the A reads so the output is already "transposed" for the msg combine.
// Block = 256 threads = 8 waves; each wave computes a 16-row x 64-col tile
// via V_WMMA_F32_16X16X4_F32, K swept in steps of 4 (zero-padded to KPAD).
// W is staged in LDS with K-pairs interleaved so each B fragment is a single
// aligned 8-byte ds_load into an even VGPR pair (no register repacking).
// ---------------------------------------------------------------------------
template<int KPAD>
__global__ __launch_bounds__(256) void gemm_wmma_k(
    const float* __restrict__ Ain, const float* __restrict__ W,
    float* __restrict__ Out, int M, int K, int C, int relu, int remap)
{
  __shared__ float As[128 * KPAD];
  __shared__ float Bs[KPAD * 64];   // layout: ((k>>1)*64 + c)*2 + (k&1)
  const int tid = threadIdx.x;
  const int rowBase = blockIdx.x * 128;

  // Stage W (K x C) into LDS, K-pair interleaved, zero padded to KPAD x 64.
  for (int idx = tid; idx < KPAD * 64; idx += 256) {
    int k = idx >> 6, c = idx & 63;
    float v = (k < K && c < C) ? W[k * C + c] : 0.f;
    Bs[(((k >> 1) * 64) + c) * 2 + (k & 1)] = v;
  }
  // Stage 128 A rows into LDS (coalesced along K), zero padded.
  for (int idx = tid; idx < 128 * KPAD; idx += 256) {
    int r = idx / KPAD, k = idx - r * KPAD;
    int g = rowBase + r;
    float v = 0.f;
    if (g < M && k < K) {
      long phys = g;
      if (remap) {  // logical row (b,n,m) -> physical row (b,m,n)
        unsigned gu = (unsigned)g;
        unsigned bn = gu / NN; unsigned m = gu - bn * NN;
        unsigned b  = bn / NN; unsigned n = bn - b * NN;
        phys = ((long)b * NN + m) * NN + n;
      }
      v = Ain[phys * (long)K + k];
    }
    As[idx] = v;
  }
  __syncthreads();

  const int wave  = tid >> 5;
  const int lane  = tid & 31;
  const int mrow  = lane & 15;
  const int hi    = lane >> 4;      // 0/1 : lanes 16-31 hold K+2..K+3 (A) / rows +8 (C)
  const int khalf = hi * 2;
  const float* Arow = &As[(wave * 16 + mrow) * KPAD];

  v8f acc[4];
#pragma unroll
  for (int t = 0; t < 4; ++t) acc[t] = (v8f){0,0,0,0,0,0,0,0};

  for (int kk = 0; kk < KPAD; kk += 4) {
    // A 16x4 fragment: lane holds K = kk+khalf, kk+khalf+1 (contiguous, aligned)
    v2f a = *(const v2f*)&Arow[kk + khalf];
    const int p = (kk + khalf) >> 1;            // K-pair index
    const float* Bp = &Bs[(p * 64 + mrow) * 2];
#pragma unroll
    for (int t = 0; t < 4; ++t) {   // 4 column tiles of 16 (cols padded to 64)
      v2f bb = *(const v2f*)&Bp[t * 32];        // (p*64 + t*16+mrow)*2
      acc[t] = __builtin_amdgcn_wmma_f32_16x16x4_f32(
                 false, a, false, bb, (short)0, acc[t], false, false);
    }
  }

  // C/D layout: vgpr j, lanes0-15 -> row j, lanes16-31 -> row j+8; col = lane&15
  const int rb = rowBase + wave * 16 + hi * 8;
#pragma unroll
  for (int j = 0; j < 8; ++j) {
    int r = rb + j;
    if (r < M) {
#pragma unroll
      for (int t = 0; t < 4; ++t) {
        int c = t * 16 + mrow;
        if (c < C) {
          float v = acc[t][j];
          if (relu) v = fmaxf(v, 0.f);
          Out[(long)r * C + c] = v;
        }
      }
    }
  }
}

// ---------------------------------------------------------------------------
// Masked reduce over one N axis:
//   axis2==0 : S[b,n,d]  = sum_m H[b,m,n,d]*A[b,m,n]   (rows stride N*D)
//   axis2==1 : Mv[b,n,d] = sum_m H[b,n,m,d]*A[b,n,m]   (rows stride D)
// Optionally also copies X into cols [0,32) of the (stride-82) XM buffer.
// ---------------------------------------------------------------------------
__global__ void masked_reduce_k(const float* __restrict__ H,
                                const float* __restrict__ Aedge,
                                float* __restrict__ Out, int out_stride,
                                int out_off, int axis2,
                                const float* __restrict__ Xcopy)
{
  int bn = blockIdx.x;            // 0 .. BNR-1
  int b = bn / NN, n = bn - (bn / NN) * NN;
  int d = threadIdx.x;            // 64 threads
  if (d < DD) {
    float s = 0.f;
    for (int m = 0; m < NN; ++m) {
      long r = axis2 ? ((long)(b * NN + n) * NN + m)
                     : ((long)(b * NN + m) * NN + n);
      s += H[r * DD + d] * Aedge[r];
    }
    Out[(long)bn * out_stride + out_off + d] = s;
  }
  if (Xcopy && d < F1C)
    Out[(long)bn * out_stride + d] = Xcopy[(long)bn * F1C + d];
}

// ---------------------------------------------------------------------------
// Hpre[r,d] = H0[r,d] + A[r]*(T[r/N, d] - Q[r,d]);  fused BN sum/sumsq stats.
// Thread layout: d = tid&63 (channel), rg = tid>>6 walks rows; block = 256 rows.
// ---------------------------------------------------------------------------
__global__ __launch_bounds__(256) void combine_stats_k(
    const float* __restrict__ H0, const float* __restrict__ Q,
    const float* __restrict__ Aedge, const float* __restrict__ T,
    float* __restrict__ Hpre, float* __restrict__ gsum, float* __restrict__ gsq)
{
  const int tid = threadIdx.x;
  const int d  = tid & 63;
  const int rg = tid >> 6;
  const int base = blockIdx.x * 256;
  const bool act = (d < DD);
  float sum = 0.f, sq = 0.f;
  int lim = base + 256; if (lim > EE) lim = EE;
  for (int r = base + rg; r < lim; r += 4) {
    if (act) {
      int bn = r / NN;
      float a  = Aedge[r];
      float tt = T[bn * DD + d];
      long off = (long)r * DD + d;
      float v = H0[off] + a * (tt - Q[off]);
      Hpre[off] = v;
      sum += v; sq += v * v;
    }
  }
  __shared__ float ls[4 * 64];
  __shared__ float lq[4 * 64];
  ls[rg * 64 + d] = sum;
  lq[rg * 64 + d] = sq;
  __syncthreads();
  if (rg == 0 && act) {
    atomicAdd(&gsum[d], ls[d] + ls[64 + d] + ls[128 + d] + ls[192 + d]);
    atomicAdd(&gsq[d],  lq[d] + lq[64 + d] + lq[128 + d] + lq[192 + d]);
  }
}

// sum/sumsq -> per-channel (scale, shift) for y = x*scale + shift
__global__ void finalize_stats_k(const float* __restrict__ s,
                                 const float* __restrict__ q,
                                 const float* __restrict__ g,
                                 const float* __restrict__ bb,
                                 float invcount, int C,
                                 float* __restrict__ scale,
                                 float* __restrict__ shift)
{
  int d = threadIdx.x;
  if (d < C) {
    float m   = s[d] * invcount;
    float var = q[d] * invcount - m * m;
    float is  = rsqrtf(var + EPS);
    float sc  = g[d] * is;
    scale[d] = sc;
    shift[d] = bb[d] - m * sc;
  }
}

// In-place BN + ReLU over a (rows x 50) fp32 tensor, float4 vectorized.
__global__ void bn_relu_k(float* __restrict__ H,
                          const float* __restrict__ scale,
                          const float* __restrict__ shift, int total4)
{
  __shared__ float sc[DD], sh[DD];
  if (threadIdx.x < DD) { sc[threadIdx.x] = scale[threadIdx.x];
                          sh[threadIdx.x] = shift[threadIdx.x]; }
  __syncthreads();
  int stride = gridDim.x * blockDim.x;
  for (int i = blockIdx.x * blockDim.x + threadIdx.x; i < total4; i += stride) {
    long e = (long)i * 4;
    int d = (int)(e % DD);
    float4 v = reinterpret_cast<float4*>(H)[i];
    float* vp = &v.x;
#pragma unroll
    for (int k = 0; k < 4; ++k) {
      vp[k] = fmaxf(vp[k] * sc[d] + sh[d], 0.f);
      if (++d == DD) d = 0;
    }
    reinterpret_cast<float4*>(H)[i] = v;
  }
}

__global__ void zero_k(float* __restrict__ p, int n)
{ int i = blockIdx.x * blockDim.x + threadIdx.x; if (i < n) p[i] = 0.f; }

// h[b,d] = sum_n Hread[b,n,d]; hcat = [h | ESM1b] (512 x 100)
__global__ void rowsum_concat_k(const float* __restrict__ Hread,
                                const float* __restrict__ esm,
                                float* __restrict__ hcat)
{
  int b = blockIdx.x, d = threadIdx.x;     // 512 blocks x 64 threads
  if (d < DD) {
    float s = 0.f;
    for (int n = 0; n < NN; ++n) s += Hread[((long)b * NN + n) * DD + d];
    hcat[b * 100 + d]      = s;
    hcat[b * 100 + DD + d] = esm[b * DD + d];
  }
}

// BN over batch (512) for 100 channels -> scale/shift
__global__ void head_stats_k(const float* __restrict__ hcat,
                             const float* __restrict__ g,
                             const float* __restrict__ bb,
                             float* __restrict__ scale, float* __restrict__ shift)
{
  int c = threadIdx.x;                     // 128 threads
  if (c < 100) {
    float s = 0.f, q = 0.f;
    for (int b = 0; b < NB; ++b) { float v = hcat[b * 100 + c]; s += v; q += v * v; }
    float m   = s * (1.f / NB);
    float var = q * (1.f / NB) - m * m;
    float is  = rsqrtf(var + EPS);
    scale[c] = g[c] * is;
    shift[c] = bb[c] - m * g[c] * is;
  }
}

// y[b] = sigmoid( relu(bn(hcat[b]) @ W1 + bL1) @ W2 + bL2 )
__global__ void head_final_k(const float* __restrict__ hcat,
                             const float* __restrict__ scale,
                             const float* __restrict__ shift,
                             const float* __restrict__ W1,
                             const float* __restrict__ bL1,
                             const float* __restrict__ W2,
                             const float* __restrict__ bL2,
                             float* __restrict__ out)
{
  int b = blockIdx.x * blockDim.x + threadIdx.x;
  if (b >= NB) return;
  float x[100];
  for (int c = 0; c < 100; ++c) x[c] = hcat[b * 100 + c] * scale[c] + shift[c];
  float acc2 = bL2[0];
  for (int j = 0; j < 32; ++j) {
    float a = bL1[j];
    for (int c = 0; c < 100; ++c) a += x[c] * W1[c * 32 + j];
    a = fmaxf(a, 0.f);
    acc2 += a * W2[j];
  }
  out[b] = 1.f / (1.f + expf(-acc2));
}

// ---------------------------------------------------------------------------
extern "C" void kernel_launch(void* const* d_in, const int* in_sizes, int n_in,
                              void* d_out, int out_size, void* d_ws, size_t ws_size,
                              hipStream_t stream)
{
  const float* XE    = (const float*)d_in[0];
  const float* X     = (const float*)d_in[1];
  const float* A     = (const float*)d_in[2];
  const float* ESM   = (const float*)d_in[3];
  const float* Wi    = (const float*)d_in[4];
  const float* Wm1   = (const float*)d_in[5];
  const float* Wm2   = (const float*)d_in[6];
  const float* Wa    = (const float*)d_in[7];
  const float* g1    = (const float*)d_in[8];
  const float* b1    = (const float*)d_in[9];
  const float* g2    = (const float*)d_in[10];
  const float* b2    = (const float*)d_in[11];
  const float* g3    = (const float*)d_in[12];
  const float* b3    = (const float*)d_in[13];
  const float* W1    = (const float*)d_in[14];
  const float* bL1   = (const float*)d_in[15];
  const float* W2    = (const float*)d_in[16];
  const float* bL2   = (const float*)d_in[17];
  float* out = (float*)d_out;

  // workspace layout (floats); total ~1.54 GB
  float* ws   = (float*)d_ws;
  const long EDL = (long)EE * DD;          // 125,440,000
  float* H0   = ws;
  float* Hbuf = ws + EDL;                  // H1pre/H1 then H2pre/H2
  float* Q    = ws + 2 * EDL;              // (H@Wm) with (n,m) swapped
  float* aux  = ws + 3 * EDL;
  float* Sbuf  = aux;                       long o = (long)BNR * DD;
  float* Tbuf  = aux + o;                   o += (long)BNR * DD;
  float* XM    = aux + o;                   o += (long)BNR * KXM;
  float* Hread = aux + o;                   o += (long)BNR * DD;
  float* hcat  = aux + o;                   o += (long)NB * 100;
  float* ssum  = aux + o;                   o += 64;
  float* ssq   = aux + o;                   o += 64;
  float* scl   = aux + o;                   o += 64;
  float* shf   = aux + o;                   o += 64;
  float* scl3  = aux + o;                   o += 128;
  float* shf3  = aux + o;

  const int gE  = (EE  + 127) / 128;       // 19600
  const int gBN = (BNR + 127) / 128;       // 280
  const int cmb = EE / 256;                // 9800 (EE divisible by 256)
  const int t4  = (int)(EDL / 4);          // 31,360,000
  const int gT4 = (t4 + 255) / 256;
  const float invE = 1.f / (float)EE;

  // H0 = relu(XE @ Wi)
  gemm_wmma_k<44><<<gE, 256, 0, stream>>>(XE, Wi, H0, EE, FF, DD, 1, 0);

  // ---- layer 1 ----
  masked_reduce_k<<<BNR, 64, 0, stream>>>(H0, A, Sbuf, DD, 0, 0, nullptr);
  gemm_wmma_k<52><<<gBN, 256, 0, stream>>>(Sbuf, Wm1, Tbuf, BNR, DD, DD, 0, 0);
  gemm_wmma_k<52><<<gE, 256, 0, stream>>>(H0, Wm1, Q, EE, DD, DD, 0, 1);
  zero_k<<<1, 128, 0, stream>>>(ssum, 128);   // zeroes ssum(64)+ssq(64)
  combine_stats_k<<<cmb, 256, 0, stream>>>(H0, Q, A, Tbuf, Hbuf, ssum, ssq);
  finalize_stats_k<<<1, 64, 0, stream>>>(ssum, ssq, g1, b1, invE, DD, scl, shf);
  bn_relu_k<<<gT4, 256, 0, stream>>>(Hbuf, scl, shf, t4);          // Hbuf = H1

  // ---- layer 2 ----
  masked_reduce_k<<<BNR, 64, 0, stream>>>(Hbuf, A, Sbuf, DD, 0, 0, nullptr);
  gemm_wmma_k<52><<<gBN, 256, 0, stream>>>(Sbuf, Wm2, Tbuf, BNR, DD, DD, 0, 0);
  gemm_wmma_k<52><<<gE, 256, 0, stream>>>(Hbuf, Wm2, Q, EE, DD, DD, 0, 1);
  zero_k<<<1, 128, 0, stream>>>(ssum, 128);
  combine_stats_k<<<cmb, 256, 0, stream>>>(H0, Q, A, Tbuf, Hbuf, ssum, ssq);
  finalize_stats_k<<<1, 64, 0, stream>>>(ssum, ssq, g2, b2, invE, DD, scl, shf);
  bn_relu_k<<<gT4, 256, 0, stream>>>(Hbuf, scl, shf, t4);          // Hbuf = H2

  // ---- readout ----
  masked_reduce_k<<<BNR, 64, 0, stream>>>(Hbuf, A, XM, KXM, F1C, 1, X);  // XM=[X|Mv]
  gemm_wmma_k<84><<<gBN, 256, 0, stream>>>(XM, Wa, Hread, BNR, KXM, DD, 1, 0);
  rowsum_concat_k<<<NB, 64, 0, stream>>>(Hread, ESM, hcat);
  head_stats_k<<<1, 128, 0, stream>>>(hcat, g3, b3, scl3, shf3);
  head_final_k<<<2, 256, 0, stream>>>(hcat, scl3, shf3, W1, bL1, W2, bL2, out);
}